// CustomMaskRCNN_47175920779771
// MI455X (gfx1250) — compile-verified
//
#include <hip/hip_runtime.h>
#include <math.h>

#define P_N 512
#define G_N 200
#define MASK_H 520
#define MASK_W 704
#define MS 28

typedef __attribute__((ext_vector_type(2))) float v2f;
typedef __attribute__((ext_vector_type(8))) float v8f;

// ---------------------------------------------------------------------------
// Kernel 1: per-proposal max-IoU / argmax over 200 GT boxes (GT staged in LDS)
// ---------------------------------------------------------------------------
__global__ __launch_bounds__(512) void iou_kernel(const float* __restrict__ proposals,
                                                  const float* __restrict__ gt_boxes,
                                                  float* __restrict__ pos_out,
                                                  int* __restrict__ midx_out) {
    __shared__ float gb[G_N * 4];
    int tid = threadIdx.x;
    for (int i = tid; i < G_N * 4; i += blockDim.x) gb[i] = gt_boxes[i];
    __syncthreads();
    int p = tid;
    if (p >= P_N) return;
    float ax1 = proposals[p * 4 + 0], ay1 = proposals[p * 4 + 1];
    float ax2 = proposals[p * 4 + 2], ay2 = proposals[p * 4 + 3];
    float area_a = (ax2 - ax1) * (ay2 - ay1);
    float best = -1.0f;
    int bidx = 0;
    for (int g = 0; g < G_N; ++g) {
        float bx1 = gb[g * 4 + 0], by1 = gb[g * 4 + 1];
        float bx2 = gb[g * 4 + 2], by2 = gb[g * 4 + 3];
        float area_b = (bx2 - bx1) * (by2 - by1);
        float lx = fmaxf(ax1, bx1), ly = fmaxf(ay1, by1);
        float rx = fminf(ax2, bx2), ry = fminf(ay2, by2);
        float w = fmaxf(rx - lx, 0.0f), h = fmaxf(ry - ly, 0.0f);
        float inter = w * h;
        float iou = inter / (area_a + area_b - inter);
        if (iou > best) { best = iou; bidx = g; }  // strict > keeps first argmax
    }
    pos_out[p] = (best > 0.3f) ? 1.0f : 0.0f;
    midx_out[p] = bidx;
}

// ---------------------------------------------------------------------------
// Kernel 2: one block per proposal — bilinear target gather + BCE, block sum.
// Early-out for non-positive proposals (their contribution is zeroed anyway).
// ---------------------------------------------------------------------------
__global__ __launch_bounds__(256) void bce_kernel(const float* __restrict__ mask_logits,
                                                  const float* __restrict__ gt_boxes,
                                                  const float* __restrict__ gt_masks,
                                                  const float* __restrict__ pos_in,
                                                  const int* __restrict__ midx_in,
                                                  float* __restrict__ bce_sums) {
    int p = blockIdx.x;
    int tid = threadIdx.x;
    __shared__ float sdata[256];

    if (pos_in[p] == 0.0f) {
        if (tid == 0) bce_sums[p] = 0.0f;
        return;
    }
    int g = midx_in[p];
    // truncate-toward-zero cast matches jnp .astype(int32) for these positive boxes
    int b0 = (int)gt_boxes[g * 4 + 0];
    int b1 = (int)gt_boxes[g * 4 + 1];
    int b2 = (int)gt_boxes[g * 4 + 2];
    int b3 = (int)gt_boxes[g * 4 + 3];
    int x1 = min(max(b0, 0), MASK_W - 1);
    int y1 = min(max(b1, 0), MASK_H - 1);
    int x2 = max(x1 + 1, min(b2, MASK_W));
    int y2 = max(y1 + 1, min(b3, MASK_H));
    int cwi = x2 - x1, chi = y2 - y1;
    float cw = (float)cwi, ch = (float)chi;

    const float* mrow = gt_masks + (size_t)g * MASK_H * MASK_W;
    const float* lrow = mask_logits + ((size_t)p * 2 + 1) * (MS * MS);

    float acc = 0.0f;
    for (int s = tid; s < MS * MS; s += 256) {
        int i = s / MS;  // y index
        int j = s % MS;  // x index
        float sy = ((float)i + 0.5f) * ch * (1.0f / MS) - 0.5f;
        sy = fminf(fmaxf(sy, 0.0f), ch - 1.0f);
        float sx = ((float)j + 0.5f) * cw * (1.0f / MS) - 0.5f;
        sx = fminf(fmaxf(sx, 0.0f), cw - 1.0f);
        int y0 = (int)floorf(sy);
        int x0 = (int)floorf(sx);
        int yp = min(y0 + 1, chi - 1);
        int xp = min(x0 + 1, cwi - 1);
        float wy = sy - (float)y0;
        float wx = sx - (float)x0;
        int Y0 = y1 + y0, Y1 = y1 + yp, X0 = x1 + x0, X1 = x1 + xp;
        float t00 = mrow[Y0 * MASK_W + X0];
        float t01 = mrow[Y0 * MASK_W + X1];
        float t10 = mrow[Y1 * MASK_W + X0];
        float t11 = mrow[Y1 * MASK_W + X1];
        float tgt = (1.0f - wy) * (1.0f - wx) * t00 + (1.0f - wy) * wx * t01 +
                    wy * (1.0f - wx) * t10 + wy * wx * t11;
        float x = lrow[s];
        float sp = fmaxf(x, 0.0f) + log1pf(expf(-fabsf(x)));  // stable softplus
        acc += sp - x * tgt;
    }
    sdata[tid] = acc;
    __syncthreads();
    for (int stride = 128; stride > 0; stride >>= 1) {
        if (tid < stride) sdata[tid] += sdata[tid + stride];
        __syncthreads();
    }
    if (tid == 0) bce_sums[p] = sdata[0];
}

// ---------------------------------------------------------------------------
// Kernel 3: single wave32. num_pos via shuffle reduction; total BCE via
// V_WMMA_F32_16X16X4_F32 with all-ones B (row-sum accumulation in f32 C).
// A layout (32-bit 16x4): lanes 0-15 hold (M=lane, K=0/1), lanes 16-31 hold
// (M=lane-16, K=2/3). Chunk stored row-major: elem(M,K) = chunk[M*4+K].
// After 8 chunks every column of D holds the accumulated row sums; lane 0
// (M=0..7, N=0) + lane 16 (M=8..15, N=0) give the exact total.
// ---------------------------------------------------------------------------
__global__ __launch_bounds__(32) void finalize_kernel(const float* __restrict__ bce_sums,
                                                      const float* __restrict__ pos_in,
                                                      float* __restrict__ out) {
    int lane = threadIdx.x;

    float np = 0.0f;
    for (int i = lane; i < P_N; i += 32) np += pos_in[i];
    for (int off = 16; off > 0; off >>= 1) np += __shfl_xor(np, off, 32);

    v8f c = {};
    v2f bmat;
    bmat.x = 1.0f;
    bmat.y = 1.0f;
    int m = lane & 15;
    int kbase = (lane < 16) ? 0 : 2;
    for (int chunk = 0; chunk < P_N / 64; ++chunk) {
        const float* base = bce_sums + chunk * 64;
        v2f a;
        a.x = base[m * 4 + kbase];
        a.y = base[m * 4 + kbase + 1];
        c = __builtin_amdgcn_wmma_f32_16x16x4_f32(
            /*neg_a=*/false, a, /*neg_b=*/false, bmat,
            /*c_mod=*/(short)0, c, /*reuse_a=*/false, /*reuse_b=*/false);
    }
    float local = c[0] + c[1] + c[2] + c[3] + c[4] + c[5] + c[6] + c[7];
    float total = local + __shfl_xor(local, 16, 32);

    if (lane == 0) {
        float denom = fmaxf(np, 1.0f) * (float)(MS * MS);
        out[0] = (np > 0.0f) ? (total / denom) : 0.0f;
    }
}

extern "C" void kernel_launch(void* const* d_in, const int* in_sizes, int n_in,
                              void* d_out, int out_size, void* d_ws, size_t ws_size,
                              hipStream_t stream) {
    (void)in_sizes; (void)n_in; (void)out_size; (void)ws_size;
    const float* mask_logits = (const float*)d_in[0];
    const float* proposals   = (const float*)d_in[1];
    const float* gt_boxes    = (const float*)d_in[2];
    const float* gt_masks    = (const float*)d_in[3];
    // d_in[4] = gt_labels: unused by the reference computation

    float* ws       = (float*)d_ws;
    float* bce_sums = ws;              // [0,   512)
    float* pos      = ws + P_N;        // [512, 1024)
    int*   midx     = (int*)(ws + 2 * P_N);  // [1024, 1536)
    float* out      = (float*)d_out;

    iou_kernel<<<1, 512, 0, stream>>>(proposals, gt_boxes, pos, midx);
    bce_kernel<<<P_N, 256, 0, stream>>>(mask_logits, gt_boxes, gt_masks, pos, midx, bce_sums);
    finalize_kernel<<<1, 32, 0, stream>>>(bce_sums, pos, out);
}